// MessagePassing_46926812676142
// MI455X (gfx1250) — compile-verified
//
#include <hip/hip_runtime.h>

// CDNA5 / gfx1250: wave32, WMMA f32 16x16x4 (fp32 matrix pipe).
typedef float v2f __attribute__((ext_vector_type(2)));
typedef float v8f __attribute__((ext_vector_type(8)));

#define DSZ   96
#define PLANE (96 * 96)
#define VOL   (96 * 96 * 96)
#define NCH   17

// One block stages a zero-padded input tile (WPB d-planes + halo) x (16+halo) x (NC*4)
// into LDS, plus zero-expanded Toeplitz weight rows. Each wave32 then computes one
// 16x16 (H x W) output tile at its own depth d = d0 + wave, accumulating all K^2
// tap-planes in the WMMA C registers (preloaded with dst for the "+=").
template <int K, int DIL, int WPB>
__global__ __launch_bounds__(WPB * 32) void conv_acc(float* __restrict__ buf,
                                                     const float* __restrict__ wgt,
                                                     int srcC, int dstC) {
    constexpr int CTR   = K / 2;                  // center tap
    constexpr int HALO  = DIL * (K - 1);          // total halo per axis
    constexpr int L     = 16 + HALO;              // extended width/height per tile
    constexpr int NC    = (L + 3) / 4;            // K=4 chunks per 1D conv
    constexpr int NCP   = NC * 4;                 // padded LDS row length (even)
    constexpr int NP    = WPB + HALO;             // staged depth planes
    constexpr int EXT   = NCP + 16;               // expanded Toeplitz row length
    constexpr int BLOCK = WPB * 32;

    __shared__ float tile[NP * L * NCP];          // zero-padded input tile
    __shared__ float ext[K * K * EXT];            // zero-expanded Toeplitz rows

    int tid = threadIdx.x;
    int t = blockIdx.x;
    int wt = t % 6;  t /= 6;
    int ht = t % 6;  t /= 6;
    int db = t % (DSZ / WPB); t /= (DSZ / WPB);
    int b  = t;

    int d0 = db * WPB, h0 = ht * 16, w0 = wt * 16;

    const float* __restrict__ src = buf + (size_t)(b * NCH + srcC) * VOL;
    float*       __restrict__ dst = buf + (size_t)(b * NCH + dstC) * VOL;

    // ---- Stage input tile with zeroed halo (all boundary masking happens here) ----
    constexpr int TOT = NP * L * NCP;
    for (int i = tid; i < TOT; i += BLOCK) {
        int pl = i / (L * NCP);
        int r2 = i - pl * (L * NCP);
        int ro = r2 / NCP;
        int co = r2 - ro * NCP;
        int zd = d0 - DIL * CTR + pl;
        int yh = h0 - DIL * CTR + ro;
        int wx = w0 - DIL * CTR + co;
        bool ok = ((unsigned)zd < (unsigned)DSZ) && ((unsigned)yh < (unsigned)DSZ) &&
                  ((unsigned)wx < (unsigned)DSZ);
        int off = ok ? (zd * PLANE + yh * DSZ + wx) : 0;
        float v = src[off];
        tile[i] = ok ? v : 0.0f;
    }

    // ---- Expand Toeplitz rows: ext[t][diff+15] = wrow_t[diff/DIL] (0 outside band) ----
    for (int i = tid; i < K * K * EXT; i += BLOCK) {
        int tt = i / EXT;
        int j  = i - tt * EXT;
        int diff = j - 15;
        bool ok = ((unsigned)diff < (unsigned)(K * DIL)) && (DIL == 1 || ((diff & 1) == 0));
        int wi = ok ? (tt * K + (DIL == 2 ? (diff >> 1) : diff)) : 0;
        float v = wgt[wi];
        ext[i] = ok ? v : 0.0f;
    }
    __syncthreads();

    int lane = tid & 31;
    int wave = tid >> 5;
    int m  = lane & 15;                            // A row (M)
    int kk = (lane >> 4) * 2;                      // K sub-index base for A/B fragments
    int n  = lane & 15;                            // B/C/D column (N)
    int dz = d0 + wave;                            // this wave's output depth

    // Preload C with current destination tile -> "+=" for free.
    // C/D layout: VGPR r, lanes 0-15: M=r; lanes 16-31: M=r+8; N=lane&15.
    v8f c;
#pragma unroll
    for (int r = 0; r < 8; ++r) {
        int mm = r + (lane >> 4) * 8;
        c[r] = dst[(size_t)dz * PLANE + (size_t)(h0 + mm) * DSZ + (w0 + n)];
    }

    for (int kd = 0; kd < K; ++kd) {
        for (int kh = 0; kh < K; ++kh) {
            // A fragments: consecutive, 8B-aligned LDS pair -> ds_load_b64
            const float* __restrict__ ap =
                &tile[((wave + DIL * kd) * L + m + DIL * kh) * NCP + kk];
            // B fragments: unconditional pair from expanded Toeplitz row -> ds_load_2addr
            const float* __restrict__ ep =
                &ext[(kd * K + kh) * EXT + 15 - n + kk];
#pragma unroll
            for (int cc = 0; cc < NC; ++cc) {
                v2f a = *(const v2f*)(ap + 4 * cc);
                v2f bfrag;
                bfrag.x = ep[4 * cc];
                bfrag.y = ep[4 * cc + 1];
                // 8 args: (neg_a, A, neg_b, B, c_mod, C, reuse_a, reuse_b)
                c = __builtin_amdgcn_wmma_f32_16x16x4_f32(
                        false, a, false, bfrag, (short)0, c, false, false);
            }
        }
    }

#pragma unroll
    for (int r = 0; r < 8; ++r) {
        int mm = r + (lane >> 4) * 8;
        dst[(size_t)dz * PLANE + (size_t)(h0 + mm) * DSZ + (w0 + n)] = c[r];
    }
}

extern "C" void kernel_launch(void* const* d_in, const int* in_sizes, int n_in,
                              void* d_out, int out_size, void* d_ws, size_t ws_size,
                              hipStream_t stream) {
    const float* feature = (const float*)d_in[0];
    float* out = (float*)d_out;

    // Seed output with all 17 channels (covers untouched channels 0,1,14,15,16).
    hipMemcpyAsync(out, feature, (size_t)out_size * sizeof(float),
                   hipMemcpyDeviceToDevice, stream);

    auto W = [&](int i) { return (const float*)d_in[i]; };

    dim3 blk8(256), blk4(128);
    dim3 g8(2 * (DSZ / 8) * 6 * 6);    // 864 blocks, 8 waves (8 depths) each
    dim3 g4(2 * (DSZ / 4) * 6 * 6);    // 1728 blocks, 4 waves each (dil=2, LDS budget)

    // DAG order = stream order; every edge has src != dst channel.
    conv_acc<7, 1, 8><<<g8, blk8, 0, stream>>>(out, W(1),  0, 4);    // f4  += conv(f0,  w04)
    conv_acc<7, 1, 8><<<g8, blk8, 0, stream>>>(out, W(2),  0, 5);    // f5  += conv(f0,  w05)
    conv_acc<3, 1, 8><<<g8, blk8, 0, stream>>>(out, W(3),  5, 2);    // f2  += conv(f5,  w52)
    conv_acc<7, 1, 8><<<g8, blk8, 0, stream>>>(out, W(4),  2, 4);    // f4  += conv(f2,  w24)
    conv_acc<7, 1, 8><<<g8, blk8, 0, stream>>>(out, W(5),  1, 6);    // f6  += conv(f1,  w16)
    conv_acc<7, 1, 8><<<g8, blk8, 0, stream>>>(out, W(6),  1, 7);    // f7  += conv(f1,  w17)
    conv_acc<3, 1, 8><<<g8, blk8, 0, stream>>>(out, W(7),  7, 3);    // f3  += conv(f7,  w73)
    conv_acc<7, 1, 8><<<g8, blk8, 0, stream>>>(out, W(8),  3, 6);    // f6  += conv(f3,  w36)
    conv_acc<7, 2, 4><<<g4, blk4, 0, stream>>>(out, W(9),  2, 9);    // f9  += conv(f2,  w29,  dil=2)
    conv_acc<7, 2, 4><<<g4, blk4, 0, stream>>>(out, W(10), 3, 11);   // f11 += conv(f3,  w311, dil=2)
    conv_acc<3, 1, 8><<<g8, blk8, 0, stream>>>(out, W(11), 10, 8);   // f8  += conv(f10, w80)
    conv_acc<3, 1, 8><<<g8, blk8, 0, stream>>>(out, W(12), 8, 10);   // f10 += conv(f8,  w100)
    conv_acc<7, 1, 8><<<g8, blk8, 0, stream>>>(out, W(13), 8, 12);   // f12 += conv(f8,  w120)
    conv_acc<7, 1, 8><<<g8, blk8, 0, stream>>>(out, W(14), 10, 13);  // f13 += conv(f10, w130)
}